// RelativeAdditiveAttention_69020124447485
// MI455X (gfx1250) — compile-verified
//
#include <hip/hip_runtime.h>
#include <math.h>

typedef __attribute__((ext_vector_type(2))) float v2f;
typedef __attribute__((ext_vector_type(8))) float v8f;

#define Nn 255          // neighbors per agent (T-1)
#define Kd 128          // feature dim
#define XPITCH 132      // LDS row pitch (floats): avoids 64-bank conflicts, keeps 8B align

__global__ __launch_bounds__(256)
void raa_kernel(const float* __restrict__ qx,
                const float* __restrict__ kvx,
                const float* __restrict__ Wk,
                const float* __restrict__ Wq,
                const float* __restrict__ Wv,
                const float* __restrict__ bias,
                const float* __restrict__ sw,
                float* __restrict__ out)
{
    __shared__ float s_q[Kd];
    __shared__ float s_qp[Kd];
    __shared__ float s_score[256];
    __shared__ float s_x[16 * XPITCH];
    __shared__ float s_red[16];
    __shared__ float s_t[Kd];

    const int bt   = blockIdx.x;
    const int tid  = threadIdx.x;
    const int lane = tid & 31;
    const int wave = tid >> 5;
    const float* xbase = kvx + (size_t)bt * Nn * Kd;

    // stage query row; zero score accumulators
    if (tid < Kd) s_q[tid] = qx[bt * Kd + tid];
    s_score[tid] = 0.0f;
    __syncthreads();

    // qproj[k] = q_x . Wq[:,k] + bias[k]   (coalesced over k)
    if (tid < Kd) {
        float acc = bias[tid];
        #pragma unroll 8
        for (int q = 0; q < Kd; ++q) acc = fmaf(s_q[q], Wq[q * Kd + tid], acc);
        s_qp[tid] = acc;
    }

    // This wave's B fragments of Wk for column tile j0 (4x16 f32 per k-step).
    // f32 B layout guess (A holds k0+2h, k0+2h+1): lane = column (l&15); VGPR0 rows {k0, k0+1}
    // selected by half-wave, VGPR1 rows {k0+2, k0+3}.
    const int n  = lane & 15;
    const int h  = lane >> 4;
    const int j0 = wave * 16;

    v2f bk[32];
    #pragma unroll
    for (int k0 = 0; k0 < 32; ++k0) {
        bk[k0].x = Wk[(size_t)(4 * k0 + h    ) * Kd + j0 + n];
        bk[k0].y = Wk[(size_t)(4 * k0 + h + 2) * Kd + j0 + n];
    }
    __syncthreads();              // qproj ready

    const float swl = sw[j0 + n];
    const float qpl = s_qp[j0 + n];

    // ---- Pass 1: keys GEMM via WMMA + tanh-score reduction -----------------
    for (int mt = 0; mt < 16; ++mt) {
        const int m0 = mt * 16;
        {   // cooperative staging of X[m0:m0+16, 0:128] into LDS (coalesced b128)
            const int r  = tid >> 4;
            const int c4 = (tid & 15) * 8;
            const int gr = m0 + r;
            float4 a0 = make_float4(0.f, 0.f, 0.f, 0.f), a1 = a0;
            if (gr < Nn) {
                const float* src = xbase + (size_t)gr * Kd + c4;
                a0 = *(const float4*)(src);
                a1 = *(const float4*)(src + 4);
            }
            *(float4*)&s_x[r * XPITCH + c4]     = a0;
            *(float4*)&s_x[r * XPITCH + c4 + 4] = a1;
        }
        __syncthreads();

        // 16x16 keys tile: accumulate 32 chained f32 WMMAs over K=128
        v8f c = {};
        const float* arow = &s_x[n * XPITCH + h * 2];
        #pragma unroll
        for (int k0 = 0; k0 < 32; ++k0) {
            v2f a = *(const v2f*)(arow + k0 * 4);   // ds_load_b64
            c = __builtin_amdgcn_wmma_f32_16x16x4_f32(
                    false, a, false, bk[k0], (short)0, c, false, false);
        }

        // score contribution: lane holds col (j0+n), rows m0+8h+r (r=0..7)
        #pragma unroll
        for (int r = 0; r < 8; ++r) {
            float v = swl * tanhf(c[r] + qpl);
            v += __shfl_xor(v, 1, 32);
            v += __shfl_xor(v, 2, 32);
            v += __shfl_xor(v, 4, 32);
            v += __shfl_xor(v, 8, 32);              // 16-lane row sum
            if (n == 0) atomicAdd(&s_score[m0 + h * 8 + r], v);  // ds_add_f32
        }
        __syncthreads();    // protect s_x and s_score for next tile
    }

    // ---- Softmax over 255 scores (entry 255 is padding) --------------------
    if (tid == Nn) s_score[Nn] = -3.0e38f;
    __syncthreads();
    const float sc = s_score[tid];
    float m = sc;
    #pragma unroll
    for (int mask = 16; mask >= 1; mask >>= 1) m = fmaxf(m, __shfl_xor(m, mask, 32));
    if (lane == 0) s_red[wave] = m;
    __syncthreads();
    if (tid == 0) {
        float mm = s_red[0];
        for (int i = 1; i < 8; ++i) mm = fmaxf(mm, s_red[i]);
        s_red[8] = mm;
    }
    __syncthreads();
    const float gmax = s_red[8];
    float e = __expf(sc - gmax);
    if (tid == Nn) e = 0.0f;
    float s = e;
    #pragma unroll
    for (int mask = 16; mask >= 1; mask >>= 1) s += __shfl_xor(s, mask, 32);
    if (lane == 0) s_red[wave] = s;
    __syncthreads();
    if (tid == 0) {
        float ss = 0.0f;
        for (int i = 0; i < 8; ++i) ss += s_red[i];
        s_red[9] = 1.0f / ss;
    }
    __syncthreads();
    s_score[tid] = e * s_red[9];        // s_score now holds softmax weights w
    __syncthreads();

    // ---- Pass 2: t = w^T X  (out = t . Wv removes the big values GEMM) -----
    {
        const int k    = tid & 127;
        const int half = tid >> 7;
        const int n0   = half * 128;
        const int n1   = half ? Nn : 128;
        float acc = 0.0f;
        for (int nn = n0; nn < n1; ++nn)
            acc = fmaf(s_score[nn], xbase[(size_t)nn * Kd + k], acc);
        if (half == 0) s_t[k] = acc;
        __syncthreads();
        if (half == 1) s_t[k] += acc;
        __syncthreads();
    }

    // ---- out[k] = sum_j t[j] * Wv[j][k] ------------------------------------
    if (tid < Kd) {
        float acc = 0.0f;
        #pragma unroll 8
        for (int j = 0; j < Kd; ++j) acc = fmaf(s_t[j], Wv[j * Kd + tid], acc);
        out[bt * Kd + tid] = acc;
    }
}

extern "C" void kernel_launch(void* const* d_in, const int* in_sizes, int n_in,
                              void* d_out, int out_size, void* d_ws, size_t ws_size,
                              hipStream_t stream) {
    (void)in_sizes; (void)n_in; (void)out_size; (void)d_ws; (void)ws_size;
    const float* qx   = (const float*)d_in[0];   // (8,256,128)
    const float* kvx  = (const float*)d_in[1];   // (8,256,255,128)
    const float* Wk   = (const float*)d_in[2];   // (128,128)
    const float* Wq   = (const float*)d_in[3];   // (128,128)
    const float* Wv   = (const float*)d_in[4];   // (128,128)
    const float* bias = (const float*)d_in[5];   // (128,)
    const float* sw   = (const float*)d_in[6];   // (128,)
    // d_in[7] = score_b: uniform additive shift to all scores -> softmax-invariant, unused.
    raa_kernel<<<dim3(8 * 256), dim3(256), 0, stream>>>(
        qx, kvx, Wk, Wq, Wv, bias, sw, (float*)d_out);
}